// GraphAlign_15083925144371
// MI455X (gfx1250) — compile-verified
//
#include <hip/hip_runtime.h>
#include <hip/hip_bf16.h>

typedef __attribute__((ext_vector_type(2))) float v2f;
typedef __attribute__((ext_vector_type(4))) float v4f;
typedef __attribute__((ext_vector_type(8))) float v8f;

#define BS 2
#define CH 100
#define T_DIM 100
#define D_DIM 50
#define GPAD 112          // 7 tiles of 16
#define QDIM 2800         // 100 channels * 28 bins
#define NOUT4 7000000     // 2*2800*50*25 float4 stores

// ---------------------------------------------------------------------------
// K0: per-batch column stats over channels: xx[b,t] = sum_c x^3, colmean = mean_c x
// ---------------------------------------------------------------------------
__global__ void col_stats_kernel(const float* __restrict__ x,
                                 float* __restrict__ xx,
                                 float* __restrict__ colmean) {
  int gid = blockIdx.x * blockDim.x + threadIdx.x;
  if (gid >= BS * T_DIM) return;
  int b = gid / T_DIM, t = gid % T_DIM;
  const float* xb = x + b * CH * T_DIM;
  float s3 = 0.f, s1 = 0.f;
  for (int c = 0; c < CH; ++c) {
    float v = xb[c * T_DIM + t];
    s3 += v * v * v;
    s1 += v;
  }
  xx[gid] = s3;
  colmean[gid] = s1 * (1.0f / (float)CH);
}

// ---------------------------------------------------------------------------
// K1: Gram matrix G[b] = X^T X via V_WMMA_F32_16X16X4_F32.
// One wave (32 threads) per 16x16 output tile; K=CH=100 in steps of 4.
// fp32 fragment layouts per CDNA5 ISA 7.12.2:
//   A 16x4 : lanes 0-15 -> M=lane, VGPR0=K0,VGPR1=K1 ; lanes 16-31 -> K2,K3
//   B 4x16 : symmetric (lane = N column, same K split)
//   C/D    : 8 VGPRs, lane<16 -> M=v, lane>=16 -> M=v+8, N=lane%16
// ---------------------------------------------------------------------------
__global__ void gram_wmma_kernel(const float* __restrict__ x,
                                 float* __restrict__ G) {
  const int ti = blockIdx.x;           // row tile (t)
  const int tj = blockIdx.y;           // col tile (s)
  const int b  = blockIdx.z;
  const int lane = threadIdx.x;        // 0..31, wave32
  const int half = lane >> 4;          // 0: K pair (0,1), 1: K pair (2,3)
  const int m16  = lane & 15;

  const int colA = ti * 16 + m16;      // t index this lane supplies for A
  const int colB = tj * 16 + m16;      // s index this lane supplies for B
  const int ca = colA < T_DIM ? colA : T_DIM - 1;
  const int cb = colB < T_DIM ? colB : T_DIM - 1;
  const bool va = colA < T_DIM;
  const bool vb = colB < T_DIM;

  const float* xb = x + b * CH * T_DIM;

  v8f acc = {0.f, 0.f, 0.f, 0.f, 0.f, 0.f, 0.f, 0.f};
  for (int k0 = 0; k0 < CH; k0 += 4) {
    int kk = k0 + half * 2;
    v2f af, bf;
    float a0 = xb[kk * T_DIM + ca];
    float a1 = xb[(kk + 1) * T_DIM + ca];
    float b0 = xb[kk * T_DIM + cb];
    float b1 = xb[(kk + 1) * T_DIM + cb];
    af[0] = va ? a0 : 0.f;  af[1] = va ? a1 : 0.f;
    bf[0] = vb ? b0 : 0.f;  bf[1] = vb ? b1 : 0.f;
    acc = __builtin_amdgcn_wmma_f32_16x16x4_f32(
        /*neg_a=*/false, af, /*neg_b=*/false, bf,
        /*c_mod=*/(short)0, acc, /*reuse_a=*/false, /*reuse_b=*/false);
  }

  float* Gb = G + b * GPAD * GPAD;
  const int sCol  = tj * 16 + m16;
  const int mBase = ti * 16 + (half ? 8 : 0);
#pragma unroll
  for (int v = 0; v < 8; ++v) {
    Gb[(mBase + v) * GPAD + sCol] = acc[v];
  }
}

// ---------------------------------------------------------------------------
// K2: per row t: pd[s] = 2*G[t][s] - xx[t] - xx[s]; top-3 (ties -> lower index,
// matching lax.top_k); gather colmean at the 3 indices -> ctxval[b][t][0..2]
// ---------------------------------------------------------------------------
__global__ void topk_ctx_kernel(const float* __restrict__ G,
                                const float* __restrict__ xx,
                                const float* __restrict__ colmean,
                                float* __restrict__ ctxval) {
  int gid = blockIdx.x * blockDim.x + threadIdx.x;
  if (gid >= BS * T_DIM) return;
  int b = gid / T_DIM, t = gid % T_DIM;
  const float* Grow = G + b * GPAD * GPAD + t * GPAD;
  const float* xxb = xx + b * T_DIM;
  float xt = xxb[t];
  float v0 = -3.4e38f, v1 = -3.4e38f, v2 = -3.4e38f;
  int i0 = 0, i1 = 0, i2 = 0;
  for (int s = 0; s < T_DIM; ++s) {
    float v = 2.f * Grow[s] - xt - xxb[s];
    if (v > v0)       { v2 = v1; i2 = i1; v1 = v0; i1 = i0; v0 = v; i0 = s; }
    else if (v > v1)  { v2 = v1; i2 = i1; v1 = v;  i1 = s; }
    else if (v > v2)  { v2 = v;  i2 = s; }
  }
  const float* cm = colmean + b * T_DIM;
  ctxval[gid * 3 + 0] = cm[i0];
  ctxval[gid * 3 + 1] = cm[i1];
  ctxval[gid * 3 + 2] = cm[i2];
}

// ---------------------------------------------------------------------------
// K3: bandwidth-bound emit. Output (2, 2800, 50, 100); one float4 along t per
// thread (t innermost -> fully coalesced 512B/wave stores). id enumerates
// (b, q, d, t4) with t4 innermost, so out_flat = 4*id + i.
// q = c*28 + r ; r<12 -> inner align (L=100 over x rows), r>=12 -> ctx align
// (L=3 over ctxval).
// ---------------------------------------------------------------------------
__global__ void emit_out_kernel(const float* __restrict__ x,
                                const float* __restrict__ anchors,
                                const float* __restrict__ ctxval,
                                float* __restrict__ out) {
  int id = blockIdx.x * blockDim.x + threadIdx.x;
  if (id >= NOUT4) return;

  int t4  = id % 25;  int tmp = id / 25;
  int d   = tmp % D_DIM; tmp /= D_DIM;
  int q   = tmp % QDIM;
  int b   = tmp / QDIM;
  int c   = q / 28;
  int r   = q % 28;

  const bool isCtx = (r >= 12);
  float c0 = 0.f, c1 = 0.f, c2 = 0.f;
  const float* xrow = nullptr;
  if (isCtx) {
    const float* cv = ctxval + (b * T_DIM + c) * 3;
    c0 = cv[0]; c1 = cv[1]; c2 = cv[2];
  } else {
    xrow = x + (b * CH + c) * T_DIM;
  }
  const float rb   = (isCtx ? (float)(r - 12) : (float)r) + 0.5f;
  const float resf = isCtx ? 16.f : 12.f;
  const float Lf   = isCtx ? 3.f : 100.f;
  const int   Lm1  = isCtx ? 2 : 99;

  v4f res;
#pragma unroll
  for (int i = 0; i < 4; ++i) {
    int t = t4 * 4 + i;
    int n = (b * T_DIM + t) * D_DIM + d;
    float xmin = anchors[n * 3 + 1];
    float xmax = anchors[n * 3 + 2];
    float bw    = (xmax - xmin) / resf;      // match reference's exact division
    float coord = xmin + rb * bw;
    float val = 0.f;
    if (coord >= -1.f && coord <= Lf) {
      float lo   = floorf(coord);
      float frac = coord - lo;
      int loi = (int)lo;
      loi = loi < 0 ? 0 : (loi > Lm1 ? Lm1 : loi);
      int hii = loi + 1 > Lm1 ? Lm1 : loi + 1;
      float vlo, vhi;
      if (isCtx) {
        vlo = (loi == 0) ? c0 : ((loi == 1) ? c1 : c2);
        vhi = (hii == 0) ? c0 : ((hii == 1) ? c1 : c2);
      } else {
        vlo = xrow[loi];
        vhi = xrow[hii];
      }
      val = vlo * (1.f - frac) + vhi * frac;
    }
    res[i] = val;
  }
  *(v4f*)(out + (size_t)id * 4) = res;
}

// ---------------------------------------------------------------------------
extern "C" void kernel_launch(void* const* d_in, const int* in_sizes, int n_in,
                              void* d_out, int out_size, void* d_ws, size_t ws_size,
                              hipStream_t stream) {
  const float* x       = (const float*)d_in[0];   // (2,100,100) f32
  const float* anchors = (const float*)d_in[1];   // (10000,3) f32
  // d_in[2] (index) unused by the reference.
  float* out = (float*)d_out;                     // (2,2800,50,100) f32

  float* ws      = (float*)d_ws;
  float* xx      = ws;                 // 200 floats
  float* colmean = ws + 200;           // 200 floats
  float* ctxval  = ws + 400;           // 600 floats
  float* G       = ws + 1024;          // 2*112*112 = 25088 floats

  col_stats_kernel<<<dim3(2), 128, 0, stream>>>(x, xx, colmean);
  gram_wmma_kernel<<<dim3(7, 7, 2), 32, 0, stream>>>(x, G);
  topk_ctx_kernel<<<dim3(2), 128, 0, stream>>>(G, xx, colmean, ctxval);
  emit_out_kernel<<<dim3((NOUT4 + 255) / 256), 256, 0, stream>>>(x, anchors, ctxval, out);
}